// CnnKF_39436389712505
// MI455X (gfx1250) — compile-verified
//
#include <hip/hip_runtime.h>

typedef __attribute__((ext_vector_type(2))) float v2f;
typedef __attribute__((ext_vector_type(8))) float v8f;

#define B_N 2048
#define R_N 32
#define O_N 64
#define K_N (R_N * O_N)   // 2048 = flattened contraction length (k = i*32 + l)

// One block per batch. 128 threads = 4 waves; wave w owns output columns
// o in [16w, 16w+16). Each wave sweeps all 2048 k-rows in chunks of 4 using
// V_WMMA_F32_16X16X4_F32:
//   B tile (4x16)  = obs rows [kb..kb+3], cols [16w..16w+15]   (global, NT)
//   A tile (16x4)  = v[kb..kb+3] broadcast to all 16 rows      (LDS)
//   D accumulates the partial dot product, identical in every row; row M=0
//   (acc[0], lanes 0..15) holds out[b, 16w + lane] at the end.
__global__ __launch_bounds__(128) void cnnkf_wmma_kernel(
    const float* __restrict__ ctx,   // [B, R, O]
    const float* __restrict__ obs,   // [B, O, R, O]
    float* __restrict__ out)         // [B, O]
{
    __shared__ float vlds[K_N];      // v[k] = ctx[b, R-1-l, i], k = i*R + l  (8 KB)

    const int b    = blockIdx.x;
    const int tid  = threadIdx.x;
    const int wave = tid >> 5;
    const int lane = tid & 31;

    const float* ctx_b = ctx + (size_t)b * (R_N * O_N);
    const float* obs_b = obs + (size_t)b * ((size_t)O_N * R_N * O_N);

    // Stage flipped context into LDS in k-order (coalesced global reads).
    #pragma unroll
    for (int t = 0; t < K_N / 128; ++t) {
        int idx = tid + t * 128;
        int r = idx >> 6;            // ctx row, 0..31
        int i = idx & 63;            // ctx col (contiguous -> coalesced)
        vlds[i * R_N + (R_N - 1 - r)] = ctx_b[idx];
    }
    __syncthreads();

    const int half = lane >> 4;               // 0: lanes 0-15, 1: lanes 16-31
    const int col  = (wave << 4) + (lane & 15);

    // B layout (4x16 f32): VGPR0 = row K=2*half, VGPR1 = row K=2*half+1, col = lane%16.
    const float* pB = obs_b + (size_t)(half * 2) * O_N + col;
    // A layout (16x4 f32): VGPR0 = v[kb+2*half], VGPR1 = v[kb+2*half+1], all rows equal.
    const int aBase = half * 2;

    v8f acc = {0.f, 0.f, 0.f, 0.f, 0.f, 0.f, 0.f, 0.f};

    #pragma unroll 8
    for (int kb = 0; kb < K_N; kb += 4) {
        v2f Bm, Am;
        Bm.x = __builtin_nontemporal_load(pB + (size_t)kb * O_N);        // row kb+2h
        Bm.y = __builtin_nontemporal_load(pB + (size_t)kb * O_N + O_N);  // row kb+2h+1
        Am.x = vlds[aBase + kb];
        Am.y = vlds[aBase + kb + 1];
        acc = __builtin_amdgcn_wmma_f32_16x16x4_f32(
            /*neg_a=*/false, Am, /*neg_b=*/false, Bm,
            /*c_mod=*/(short)0, acc, /*reuse_a=*/false, /*reuse_b=*/false);
    }

    // Row M=0 of D: acc[0] on lanes 0..15.
    if (lane < 16) {
        out[(size_t)b * O_N + col] = acc[0];
    }
}

extern "C" void kernel_launch(void* const* d_in, const int* in_sizes, int n_in,
                              void* d_out, int out_size, void* d_ws, size_t ws_size,
                              hipStream_t stream) {
    const float* ctx = (const float*)d_in[0];   // context: B*R*O floats
    const float* obs = (const float*)d_in[1];   // observation_IR: B*O*R*O floats
    float* out = (float*)d_out;                 // B*O floats
    (void)in_sizes; (void)n_in; (void)out_size; (void)d_ws; (void)ws_size;

    dim3 grid(B_N);
    dim3 block(128);
    hipLaunchKernelGGL(cnnkf_wmma_kernel, grid, block, 0, stream, ctx, obs, out);
}